// GATLayer_25177098289471
// MI455X (gfx1250) — compile-verified
//
#include <hip/hip_runtime.h>

#define N_NODES   50000
#define N_EDGES   800000
#define IN_DIM    128
#define NUM_HEADS 8
#define HEAD_DIM  16
#define OUT_DIM   128       // NUM_HEADS * HEAD_DIM
#define NEG_SLOPE 0.2f

typedef __attribute__((ext_vector_type(2))) float v2f;
typedef __attribute__((ext_vector_type(8))) float v8f;

// ---- order-preserving float <-> unsigned key (for integer atomicMax segment-max) ----
__device__ __forceinline__ unsigned f2key(float f) {
    unsigned b = __float_as_uint(f);
    return (b & 0x80000000u) ? ~b : (b | 0x80000000u);
}
__device__ __forceinline__ float key2f(unsigned k) {
    return (k & 0x80000000u) ? __uint_as_float(k ^ 0x80000000u)
                             : __uint_as_float(~k);
}

// =====================================================================================
// Kernel 1: fused projection GEMM (WMMA f32 16x16x4) + per-node attention scores.
// Grid: 3125 blocks x 256 threads (8 waves). Wave w of block b computes the 16x16
// output tile (node rows b*16..b*16+15) x (head w). K=128 -> 32 WMMA steps.
// =====================================================================================
__global__ __launch_bounds__(256) void gat_proj_wmma(
    const float* __restrict__ x, const float* __restrict__ W,
    const float* __restrict__ src_attn, const float* __restrict__ dst_attn,
    float* __restrict__ h, float* __restrict__ s_src, float* __restrict__ s_dst)
{
    const int wave = threadIdx.x >> 5;     // head tile 0..7
    const int lane = threadIdx.x & 31;
    const int half = lane >> 4;            // 0: K pair {0,1} / M 0..7 ; 1: {2,3} / M 8..15
    const int lm   = lane & 15;
    const int mt   = blockIdx.x;           // 0..3124 (50000 = 3125*16 exact)
    const int nt   = wave;

    const int arow = mt * 16 + lm;         // A matrix row (node)
    const int bcol = nt * 16 + lm;         // B matrix col (output channel)

    v8f c = {};
#pragma unroll
    for (int k = 0; k < IN_DIM; k += 4) {
        const int ka = k + half * 2;
        v2f a, b;
        a.x = x[arow * IN_DIM + ka];
        a.y = x[arow * IN_DIM + ka + 1];
        b.x = W[ka * OUT_DIM + bcol];
        b.y = W[(ka + 1) * OUT_DIM + bcol];
        c = __builtin_amdgcn_wmma_f32_16x16x4_f32(false, a, false, b,
                                                  (short)0, c, false, false);
    }

    // Each 16-wide N tile is exactly one head -> fuse score reduction here.
    const float sa = src_attn[nt * HEAD_DIM + lm];
    const float da = dst_attn[nt * HEAD_DIM + lm];
#pragma unroll
    for (int r = 0; r < 8; ++r) {
        const int   node = mt * 16 + r + half * 8;   // C/D layout: M = r + 8*half
        const float hv   = c[r];
        h[node * OUT_DIM + nt * HEAD_DIM + lm] = hv;

        float ts = hv * sa;
        float td = hv * da;
#pragma unroll
        for (int m = 1; m < 16; m <<= 1) {           // reduce over the 16 head dims
            ts += __shfl_xor(ts, m, 32);             // masks <16 keep halves separate
            td += __shfl_xor(td, m, 32);
        }
        if (lm == 0) {
            s_src[node * NUM_HEADS + nt] = ts;
            s_dst[node * NUM_HEADS + nt] = td;
        }
    }
}

// =====================================================================================
// Kernel 2: per-(edge,head) coefficient + leaky relu + segment-max via key atomicMax
// =====================================================================================
__global__ __launch_bounds__(256) void gat_edge_max(
    const int* __restrict__ srcIdx, const int* __restrict__ dstIdx,
    const float* __restrict__ s_src, const float* __restrict__ s_dst,
    float* __restrict__ coef_ws, unsigned* __restrict__ mkey)
{
    const int gid = blockIdx.x * blockDim.x + threadIdx.x;
    if (gid >= N_EDGES * NUM_HEADS) return;
    const int e  = gid >> 3;
    const int hd = gid & 7;
    const int s  = srcIdx[e];
    const int d  = dstIdx[e];
    float coef = s_src[s * NUM_HEADS + hd] + s_dst[d * NUM_HEADS + hd];
    coef = (coef >= 0.0f) ? coef : NEG_SLOPE * coef;
    coef_ws[gid] = coef;
    atomicMax(&mkey[d * NUM_HEADS + hd], f2key(coef));
}

// =====================================================================================
// Kernel 3: v = exp(coef - max[dst]) ; segment-sum via float atomicAdd
// =====================================================================================
__global__ __launch_bounds__(256) void gat_edge_expsum(
    const int* __restrict__ dstIdx,
    const unsigned* __restrict__ mkey,
    float* __restrict__ ev_ws, float* __restrict__ ssum)
{
    const int gid = blockIdx.x * blockDim.x + threadIdx.x;
    if (gid >= N_EDGES * NUM_HEADS) return;
    const int e  = gid >> 3;
    const int hd = gid & 7;
    const int d  = dstIdx[e];
    const float m = key2f(mkey[d * NUM_HEADS + hd]);
    const float v = expf(ev_ws[gid] - m);
    ev_ws[gid] = v;
    atomicAdd(&ssum[d * NUM_HEADS + hd], v);
}

// =====================================================================================
// Kernel 4: normalize v by segment sum
// =====================================================================================
__global__ __launch_bounds__(256) void gat_edge_norm(
    const int* __restrict__ dstIdx,
    const float* __restrict__ ssum, float* __restrict__ ev_ws)
{
    const int gid = blockIdx.x * blockDim.x + threadIdx.x;
    if (gid >= N_EDGES * NUM_HEADS) return;
    const int e  = gid >> 3;
    const int hd = gid & 7;
    const int d  = dstIdx[e];
    ev_ws[gid] = ev_ws[gid] / ssum[d * NUM_HEADS + hd];
}

// =====================================================================================
// Kernel 5: message scatter. One thread per (edge, channel): coalesced 512B gather of
// h[dst] and coalesced float atomicAdd into out[src]. All L2-resident (working set
// << 192 MB).
// =====================================================================================
__global__ __launch_bounds__(256) void gat_scatter(
    const int* __restrict__ srcIdx, const int* __restrict__ dstIdx,
    const float* __restrict__ h, const float* __restrict__ ev_ws,
    float* __restrict__ out)
{
    const int gid = blockIdx.x * blockDim.x + threadIdx.x;
    const int e   = gid >> 7;          // edge
    const int ch  = gid & 127;         // output channel
    const int hd  = ch >> 4;           // head
    if (e >= N_EDGES) return;
    const int s = srcIdx[e];
    const int d = dstIdx[e];
    const float val = h[d * OUT_DIM + ch] * ev_ws[e * NUM_HEADS + hd];
    atomicAdd(&out[s * OUT_DIM + ch], val);
}

// =====================================================================================
extern "C" void kernel_launch(void* const* d_in, const int* in_sizes, int n_in,
                              void* d_out, int out_size, void* d_ws, size_t ws_size,
                              hipStream_t stream)
{
    const float* x        = (const float*)d_in[0];
    const int*   edges    = (const int*)  d_in[1];   // [2, E] flat
    const float* W        = (const float*)d_in[2];
    const float* src_attn = (const float*)d_in[3];
    const float* dst_attn = (const float*)d_in[4];
    float*       out      = (float*)d_out;

    const int* srcIdx = edges;
    const int* dstIdx = edges + N_EDGES;

    // ---- workspace carve-up (all fp32 / u32) ----
    char* ws = (char*)d_ws;
    float*    h     = (float*)ws;                       ws += (size_t)N_NODES * OUT_DIM    * 4; // 25.6 MB
    float*    s_src = (float*)ws;                       ws += (size_t)N_NODES * NUM_HEADS  * 4; //  1.6 MB
    float*    s_dst = (float*)ws;                       ws += (size_t)N_NODES * NUM_HEADS  * 4; //  1.6 MB
    unsigned* mkey  = (unsigned*)ws;                    ws += (size_t)N_NODES * NUM_HEADS  * 4; //  1.6 MB
    float*    ssum  = (float*)ws;                       ws += (size_t)N_NODES * NUM_HEADS  * 4; //  1.6 MB
    float*    ev    = (float*)ws;                       ws += (size_t)N_EDGES * NUM_HEADS  * 4; // 25.6 MB

    // ---- init accumulators (capture-safe async memsets) ----
    hipMemsetAsync(d_out, 0, (size_t)N_NODES * OUT_DIM * 4, stream);
    hipMemsetAsync(mkey,  0, (size_t)N_NODES * NUM_HEADS * 4, stream); // key 0 == -NaN (min)
    hipMemsetAsync(ssum,  0, (size_t)N_NODES * NUM_HEADS * 4, stream);

    // ---- 1: projection + scores (WMMA) ----
    gat_proj_wmma<<<N_NODES / 16, 256, 0, stream>>>(x, W, src_attn, dst_attn,
                                                    h, s_src, s_dst);

    // ---- 2..4: edge softmax ----
    const int eh_blocks = (N_EDGES * NUM_HEADS + 255) / 256;
    gat_edge_max   <<<eh_blocks, 256, 0, stream>>>(srcIdx, dstIdx, s_src, s_dst, ev, mkey);
    gat_edge_expsum<<<eh_blocks, 256, 0, stream>>>(dstIdx, mkey, ev, ssum);
    gat_edge_norm  <<<eh_blocks, 256, 0, stream>>>(dstIdx, ssum, ev);

    // ---- 5: weighted message scatter-add ----
    const long long ec = (long long)N_EDGES * OUT_DIM;
    const int sc_blocks = (int)((ec + 255) / 256);
    gat_scatter<<<sc_blocks, 256, 0, stream>>>(srcIdx, dstIdx, h, ev, out);
}